// SSM_43224550867795
// MI455X (gfx1250) — compile-verified
//
#include <hip/hip_runtime.h>
#include <hip/hip_bf16.h>
#include <math.h>

// Shapes
#define S 16      // states
#define C 8       // chains
#define N 2000    // nodes
#define T 500     // iters
#define MPAD 512  // padded T for GEMM M-tiles

typedef float v2f __attribute__((ext_vector_type(2)));
typedef float v4f __attribute__((ext_vector_type(4)));
typedef float v8f __attribute__((ext_vector_type(8)));

// ---------------------------------------------------------------------------
// Kernel 1 (single wave32): normalize small params, run the 500-step HMM
// recursion in LINEAR space (row-stochastic transition keeps v normalized),
// emit log_hidden_state_probs and exp-space hidden probs (padded to 512 rows).
// ---------------------------------------------------------------------------
__global__ __launch_bounds__(32) void ssm_prep_kernel(
    const float* __restrict__ usi,   // [S] unnormalized state init
    const float* __restrict__ ucw,   // [S][C] unnormalized chain weights
    const float* __restrict__ ut,    // [S][S] unnormalized transition
    float* __restrict__ hid_out,     // [T][S] log hidden probs (output 3)
    float* __restrict__ Eh,          // ws [MPAD][S] exp-space hidden probs
    float* __restrict__ lcw_out)     // ws [S][C] log chain weights
{
  __shared__ float Tr[S][S];         // row-stochastic transition (linear)
  const int lane = threadIdx.x;      // 0..31 (wave32)
  const int lo16 = lane & 15;

  // --- log_softmax(chain_weights, axis=1): groups of 8 lanes, 4 states/iter
  #pragma unroll
  for (int s4 = 0; s4 < S; s4 += 4) {
    int s = s4 + (lane >> 3);
    int c = lane & 7;
    float y = ucw[s * C + c];
    float m = y;
    m = fmaxf(m, __shfl_xor(m, 4, 8));
    m = fmaxf(m, __shfl_xor(m, 2, 8));
    m = fmaxf(m, __shfl_xor(m, 1, 8));
    float e  = expf(y - m);
    float su = e;
    su += __shfl_xor(su, 4, 8);
    su += __shfl_xor(su, 2, 8);
    su += __shfl_xor(su, 1, 8);
    lcw_out[s * C + c] = y - m - logf(su);
  }

  // --- softmax(transition, axis=1) in linear space: 2 rows/iter (width-16)
  #pragma unroll
  for (int r2 = 0; r2 < S; r2 += 2) {
    int r = r2 + (lane >> 4);
    float y = ut[r * S + lo16];
    float m = y;
    m = fmaxf(m, __shfl_xor(m, 8, 16));
    m = fmaxf(m, __shfl_xor(m, 4, 16));
    m = fmaxf(m, __shfl_xor(m, 2, 16));
    m = fmaxf(m, __shfl_xor(m, 1, 16));
    float e  = expf(y - m);
    float su = e;
    su += __shfl_xor(su, 8, 16);
    su += __shfl_xor(su, 4, 16);
    su += __shfl_xor(su, 2, 16);
    su += __shfl_xor(su, 1, 16);
    Tr[r][lo16] = e / su;            // exp(log_softmax) exactly
  }
  __syncthreads();

  // Each lane caches its column of Tr in registers: tcol[i] = Tr[i][lane%16]
  float tcol[S];
  #pragma unroll
  for (int i = 0; i < S; ++i) tcol[i] = Tr[i][lo16];

  // --- log_softmax(state_init) over 16 values (full-wave reduce; upper -inf)
  float x = (lane < S) ? usi[lane] : -INFINITY;
  float m = x;
  #pragma unroll
  for (int off = 16; off >= 1; off >>= 1) m = fmaxf(m, __shfl_xor(m, off, 32));
  float e  = (lane < S) ? expf(x - m) : 0.0f;
  float su = e;
  #pragma unroll
  for (int off = 16; off >= 1; off >>= 1) su += __shfl_xor(su, off, 32);
  float lsi = x - m - logf(su);

  float v = (lane < S) ? (e / su) : 0.0f;   // linear-space hidden prob
  if (lane < S) { hid_out[lane] = lsi; Eh[lane] = v; }

  // --- 500-step recursion: v_t[j] = sum_i v_{t-1}[i] * Tr[i][j]
  // 4 split accumulators to shorten the serial FMA chain.
  for (int t = 1; t < T; ++t) {
    float a0 = 0.f, a1 = 0.f, a2 = 0.f, a3 = 0.f;
    #pragma unroll
    for (int i = 0; i < S; i += 4) {
      a0 = fmaf(__shfl(v, i + 0, 32), tcol[i + 0], a0);
      a1 = fmaf(__shfl(v, i + 1, 32), tcol[i + 1], a1);
      a2 = fmaf(__shfl(v, i + 2, 32), tcol[i + 2], a2);
      a3 = fmaf(__shfl(v, i + 3, 32), tcol[i + 3], a3);
    }
    v = (a0 + a1) + (a2 + a3);
    if (lane < S) {
      hid_out[t * S + lane] = logf(v);   // log off the critical path
      Eh[t * S + lane]      = v;
    }
  }

  // Zero-pad rows 500..511 so the GEMM M-tiles read clean data.
  for (int idx = T * S + lane; idx < MPAD * S; idx += 32) Eh[idx] = 0.0f;
}

// ---------------------------------------------------------------------------
// Kernel 2: emission log-softmax over the node axis, 1 block per state.
// Emits both log-space (for the broadcast tensor) and exp-space (for GEMM).
// ---------------------------------------------------------------------------
__global__ __launch_bounds__(256) void ssm_emission_kernel(
    const float* __restrict__ ue,    // [S][1][N]
    float* __restrict__ le,          // ws [S][N] log emission
    float* __restrict__ Ee)          // ws [S][N] exp emission
{
  __shared__ float red[256];
  const int s = blockIdx.x;
  const int tid = threadIdx.x;
  const float* row = ue + s * N;

  float m = -INFINITY;
  for (int n = tid; n < N; n += 256) m = fmaxf(m, row[n]);
  red[tid] = m; __syncthreads();
  for (int off = 128; off >= 1; off >>= 1) {
    if (tid < off) red[tid] = fmaxf(red[tid], red[tid + off]);
    __syncthreads();
  }
  m = red[0]; __syncthreads();

  float su = 0.0f;
  for (int n = tid; n < N; n += 256) su += expf(row[n] - m);
  red[tid] = su; __syncthreads();
  for (int off = 128; off >= 1; off >>= 1) {
    if (tid < off) red[tid] += red[tid + off];
    __syncthreads();
  }
  float lse = m + logf(red[0]);

  for (int n = tid; n < N; n += 256) {
    float lv = row[n] - lse;
    le[s * N + n] = lv;
    Ee[s * N + n] = expf(lv);
  }
}

// ---------------------------------------------------------------------------
// Kernel 3: the 512 MB streaming write (the roofline-dominant kernel).
// out_[t][s][c][n] = le[s][n] + lcw[s][c] + hid[t][s].  One block per (t,s);
// 128-bit non-temporal stores (TH=NT): output is written once, never re-read,
// and exceeds L2 (512 MB > 192 MB) -> keep L2 for the hot 160 KB tables.
// Uses native ext-vector v4f (HIP float4 is rejected by the NT builtin).
// ---------------------------------------------------------------------------
__global__ __launch_bounds__(256) void ssm_broadcast_kernel(
    const float* __restrict__ le,    // ws [S][N]
    const float* __restrict__ lcw,   // ws [S][C]
    const float* __restrict__ hid,   // [T][S] log hidden (already in d_out)
    float* __restrict__ out_)        // [T][S][C][N]
{
  const int blk = blockIdx.x;        // 0 .. T*S-1
  const int t = blk >> 4;
  const int s = blk & 15;

  const float h = hid[t * S + s];
  float w[C];
  #pragma unroll
  for (int c = 0; c < C; ++c) w[c] = h + lcw[s * C + c];

  const v4f* le4 = (const v4f*)(le + s * N);
  v4f* o4 = (v4f*)(out_ + (size_t)blk * (C * N));

  for (int i = threadIdx.x; i < N / 4; i += 256) {
    v4f ev = le4[i];
    #pragma unroll
    for (int c = 0; c < C; ++c) {
      v4f r = ev + w[c];   // broadcast add on native vector
      __builtin_nontemporal_store(r, &o4[c * (N / 4) + i]);
    }
  }
}

// ---------------------------------------------------------------------------
// Kernel 4: main output via f32 WMMA GEMM.
// P = Eh[500pad512 x 16] @ Ee[16 x 2000], out[t][n] = log(P[t][n]).
// One wave per 16x16 tile; K=16 -> 4x V_WMMA_F32_16X16X4_F32.
// A layout (16x4 f32): lanes 0-15 -> M, v0:K=0 / v1:K=1; lanes 16-31: K=2/K=3.
// B layout (4x16 f32): lanes 0-15 -> N, v0:K=0 / v1:K=1; lanes 16-31: K=2/K=3.
// C/D layout: VGPR r -> row r (lanes 0-15) / row r+8 (lanes 16-31).
// ---------------------------------------------------------------------------
__global__ __launch_bounds__(32) void ssm_gemm_log_kernel(
    const float* __restrict__ Eh,    // ws [MPAD][S]
    const float* __restrict__ Ee,    // ws [S][N]
    float* __restrict__ out)         // [T][N]
{
  const int tile = blockIdx.x;       // 32 M-tiles x 125 N-tiles
  const int tm = tile / (N / 16);
  const int tn = tile % (N / 16);
  const int t0 = tm * 16;
  const int n0 = tn * 16;

  const int lane = threadIdx.x;
  const int half = lane >> 4;
  const int lr   = lane & 15;

  v8f acc = {};
  #pragma unroll
  for (int kk = 0; kk < S; kk += 4) {
    const int ka = kk + 2 * half;
    v2f a, b;
    a.x = Eh[(t0 + lr) * S + ka];
    a.y = Eh[(t0 + lr) * S + ka + 1];
    b.x = Ee[ka * N + n0 + lr];
    b.y = Ee[(ka + 1) * N + n0 + lr];
    acc = __builtin_amdgcn_wmma_f32_16x16x4_f32(
        /*neg_a=*/false, a, /*neg_b=*/false, b,
        /*c_mod=*/(short)0, acc, /*reuse_a=*/false, /*reuse_b=*/false);
  }

  #pragma unroll
  for (int r = 0; r < 8; ++r) {
    const int row = t0 + r + 8 * half;
    if (row < T) {
      float lv = logf(acc[r]);
      __builtin_nontemporal_store(lv, &out[row * N + n0 + lr]);
    }
  }
}

// ---------------------------------------------------------------------------
extern "C" void kernel_launch(void* const* d_in, const int* in_sizes, int n_in,
                              void* d_out, int out_size, void* d_ws, size_t ws_size,
                              hipStream_t stream) {
  const float* usi = (const float*)d_in[0];  // [16]
  const float* ucw = (const float*)d_in[1];  // [16][8]
  const float* ue  = (const float*)d_in[2];  // [16][1][2000]
  const float* ut  = (const float*)d_in[3];  // [16][16]

  float* out      = (float*)d_out;
  float* out_main = out;                         // [500][2000]
  float* out_big  = out + (size_t)T * N;         // [500][16][8][2000]
  float* out_hid  = out_big + (size_t)T * S * C * N;  // [500][16]

  float* ws  = (float*)d_ws;
  float* Eh  = ws;                               // 512*16      = 8192 f
  float* Ee  = Eh + MPAD * S;                    // 16*2000     = 32000 f
  float* le  = Ee + S * N;                       // 16*2000     = 32000 f
  float* lcw = le + S * N;                       // 16*8        = 128 f

  ssm_prep_kernel<<<1, 32, 0, stream>>>(usi, ucw, ut, out_hid, Eh, lcw);
  ssm_emission_kernel<<<S, 256, 0, stream>>>(ue, le, Ee);
  ssm_broadcast_kernel<<<T * S, 256, 0, stream>>>(le, lcw, out_hid, out_big);
  ssm_gemm_log_kernel<<<(MPAD / 16) * (N / 16), 32, 0, stream>>>(Eh, Ee, out_main);
}